// ResBlock_15298673509110
// MI455X (gfx1250) — compile-verified
//
#include <hip/hip_runtime.h>

// ============================================================================
// Fused AdderNet residual block for MI455X (gfx1250, wave32).
//
// Roofline: 33.6 MB HBM traffic (1.44 us @ 23.3 TB/s) vs 537M f32 VALU ops
// -> VALU-bound. Sum_ci |x - w| is NOT bilinear, so WMMA cannot express it;
// the optimal inner loop is v_sub_f32 + v_add_f32 with an |src| modifier
// (abs folded into the accumulate), i.e. 2 VALU ops per element — the floor.
// CDNA5-specific paths used (matrix ops are provably inapplicable here):
//   * GLOBAL_LOAD_ASYNC_TO_LDS_B128 to stage W1|W2 (8 KB) into LDS (ASYNCcnt)
//   * s_wait_asynccnt before the workgroup barrier
//   * ds_load_b128 broadcast reads of W rows (all lanes same address)
//   * one global_prefetch per thread, lane-striped over channels, warming the
//     next workgroup's x tile in L2 (guarded: the tail block issues none)
//
// Occupancy: 1 pixel/thread, 512 blocks x 256 threads (~90 VGPRs) so every
// SIMD32 carries enough waves to hide the serial |x-w| accumulate chains.
// ============================================================================

#define CH     32
#define HWPIX  16384            // 128*128
#define NPIX   (8 * HWPIX)      // B*H*W = 131072 pixels
#define BN_EPS 1e-5f

#ifndef __has_builtin
#define __has_builtin(x) 0
#endif

#if __has_builtin(__builtin_amdgcn_global_load_async_to_lds_b128)
#define HAVE_ASYNC_LDS 1
#else
#define HAVE_ASYNC_LDS 0
#endif

// Builtin parameter types per compiler diagnostics:
//   param0: AS(1) pointer to GCC-style int vector_size(16)
typedef int v4i __attribute__((vector_size(16)));
typedef __attribute__((address_space(1))) v4i gv4i;
typedef __attribute__((address_space(3))) v4i lv4i;

// Generic->AS casts via integer round-trip (always compiles in HIP mode).
// Generic LDS pointers carry the LDS byte offset in their low 32 bits
// (ISA 10.2); generic global == AS1 bit pattern.
static __device__ __forceinline__ gv4i* as_global_v4(const void* p) {
  return (gv4i*)(unsigned long long)p;
}
static __device__ __forceinline__ lv4i* as_lds_v4(void* p) {
  return (lv4i*)(unsigned int)(unsigned long long)p;
}

static __device__ __forceinline__ void wait_async_zero() {
#if __has_builtin(__builtin_amdgcn_s_wait_asynccnt)
  __builtin_amdgcn_s_wait_asynccnt(0);
#else
  asm volatile("s_wait_asynccnt 0x0" ::: "memory");
#endif
}

// One adder layer + folded BN: yout[co] = fma(A[co], sum_ci|x-w|, B[co]).
// W row read from LDS as float4 (ds_load_b128 broadcast, conflict-free).
static __device__ __forceinline__ void adder_bn_layer(
    const float* sWlayer,          // LDS: CH*CH floats, row-major [co][ci]
    const float2* sFold,           // LDS: CH folded (A,B) pairs
    const float xin[CH], float yout[CH]) {
#pragma unroll 2
  for (int co = 0; co < CH; ++co) {
    const float4* wr = (const float4*)(sWlayer + co * CH);
    float acc0 = 0.f, acc1 = 0.f;  // two chains -> shorter dep chain
#pragma unroll
    for (int q = 0; q < CH / 8; ++q) {
      float4 wa = wr[2 * q + 0];
      float4 wb = wr[2 * q + 1];
      acc0 += fabsf(xin[8 * q + 0] - wa.x);
      acc0 += fabsf(xin[8 * q + 1] - wa.y);
      acc0 += fabsf(xin[8 * q + 2] - wa.z);
      acc0 += fabsf(xin[8 * q + 3] - wa.w);
      acc1 += fabsf(xin[8 * q + 4] - wb.x);
      acc1 += fabsf(xin[8 * q + 5] - wb.y);
      acc1 += fabsf(xin[8 * q + 6] - wb.z);
      acc1 += fabsf(xin[8 * q + 7] - wb.w);
    }
    float2 ab = sFold[co];
    yout[co] = fmaf(ab.x, acc0 + acc1, ab.y);   // BN(-sum|x-w|)
  }
}

__global__ __launch_bounds__(256) void adder_resblock_kernel(
    const float* __restrict__ x,
    const float* __restrict__ W1, const float* __restrict__ W2,
    const float* __restrict__ g1, const float* __restrict__ b1,
    const float* __restrict__ m1, const float* __restrict__ v1,
    const float* __restrict__ g2, const float* __restrict__ b2,
    const float* __restrict__ m2, const float* __restrict__ v2,
    float* __restrict__ out) {
  __shared__ __align__(16) float  sW[2 * CH * CH];   // W1 | W2, 8 KB
  __shared__ float2 sFold[2 * CH];                   // folded BN (A,B)

  const int t = threadIdx.x;

  // ---- stage W1|W2 into LDS: async global->LDS, 2 x b128 per thread ----
#if HAVE_ASYNC_LDS
  {
    const float* gsrc = (t < 128) ? (W1 + t * 8) : (W2 + (t - 128) * 8);
    float* ldst = &sW[t * 8];
    // IOFFSET applies to both the global and LDS address
    __builtin_amdgcn_global_load_async_to_lds_b128(as_global_v4(gsrc), as_lds_v4(ldst), 0, 0);
    __builtin_amdgcn_global_load_async_to_lds_b128(as_global_v4(gsrc), as_lds_v4(ldst), 16, 0);
  }
#else
  for (int i = t; i < 2 * CH * CH; i += 256)
    sW[i] = (i < CH * CH) ? W1[i] : W2[i - CH * CH];
#endif

  // ---- fold BN while the async copy is in flight ----
  // BN(-acc) = (-acc - mu)*gi + beta = fma(-gi, acc, beta - mu*gi)
  if (t < 2 * CH) {
    const int  co = t & (CH - 1);
    const bool l2 = (t >= CH);
    const float gam = l2 ? g2[co] : g1[co];
    const float bet = l2 ? b2[co] : b1[co];
    const float mu  = l2 ? m2[co] : m1[co];
    const float var = l2 ? v2[co] : v1[co];
    const float gi  = gam * rsqrtf(var + BN_EPS);
    sFold[t] = make_float2(-gi, fmaf(-mu, gi, bet));
  }

#if HAVE_ASYNC_LDS
  wait_async_zero();
#endif
  __syncthreads();

  // ---- one pixel per thread: P in [0, NPIX), 512 blocks x 256 threads ----
  const int P    = blockIdx.x * 256 + t;
  const int base = (P >> 14) * (CH * HWPIX) + (P & (HWPIX - 1));

  float xv[CH];
#pragma unroll
  for (int ci = 0; ci < CH; ++ci) xv[ci] = x[base + ci * HWPIX];

  // Cross-workgroup L2 warmup: lanes 0..31 prefetch channels 0..31 of the
  // pixel 256 ahead — together one wave covers exactly the 32 cache lines
  // the matching wave of the next block will load. The tail
  // block issues no prefetch (P2 < NPIX guard).
  {
    const int P2 = P + 256;
    if (P2 < NPIX) {
      const int nbase = (P2 >> 14) * (CH * HWPIX) + (P2 & (HWPIX - 1));
      __builtin_prefetch(&x[nbase + (t & (CH - 1)) * HWPIX], 0, 0);
    }
  }

  float hid[CH];
  adder_bn_layer(sW, sFold, xv, hid);           // layer 1 (+BN1 folded)
#pragma unroll
  for (int c = 0; c < CH; ++c) hid[c] = fmaxf(hid[c], 0.f);   // ReLU

  float y2[CH];
  adder_bn_layer(sW + CH * CH, sFold + CH, hid, y2);          // layer 2 (+BN2)

#pragma unroll
  for (int co = 0; co < CH; ++co)               // residual + ReLU, coalesced
    out[base + co * HWPIX] = fmaxf(y2[co] + xv[co], 0.f);
}

extern "C" void kernel_launch(void* const* d_in, const int* in_sizes, int n_in,
                              void* d_out, int out_size, void* d_ws, size_t ws_size,
                              hipStream_t stream) {
  (void)in_sizes; (void)n_in; (void)out_size; (void)d_ws; (void)ws_size;
  const float* x  = (const float*)d_in[0];
  const float* W1 = (const float*)d_in[1];
  const float* W2 = (const float*)d_in[2];
  const float* g1 = (const float*)d_in[3];
  const float* b1 = (const float*)d_in[4];
  const float* m1 = (const float*)d_in[5];
  const float* v1 = (const float*)d_in[6];
  const float* g2 = (const float*)d_in[7];
  const float* b2 = (const float*)d_in[8];
  const float* m2 = (const float*)d_in[9];
  const float* v2 = (const float*)d_in[10];
  float* out = (float*)d_out;

  dim3 grid(NPIX / 256), block(256);
  hipLaunchKernelGGL(adder_resblock_kernel, grid, block, 0, stream,
                     x, W1, W2, g1, b1, m1, v1, g2, b2, m2, v2, out);
}